// ProbabilisticRankingLoss_523986010405
// MI455X (gfx1250) — compile-verified
//
#include <hip/hip_runtime.h>
#include <math.h>

// Probabilistic ranking loss, N=6144 rows, 48 tasks.
// MI455X strategy: entire problem state is 96KB -> async-stage it into LDS
// (gfx1250 global_load_async_to_lds_b128 + s_wait_asynccnt), then the O(N^2)
// pairwise erf/log loop runs from LDS with conflict-free interleaved access.
// One wave per row (SPLIT=32) -> 768 blocks to fill the WGP array.
// No matmul structure -> WMMA not applicable; bound on VALU transcendentals.

#define N      6144
#define TPB    256
#define SPLIT  32         // one full wave cooperates on each row i
#define GAS __attribute__((address_space(1)))
#define LAS __attribute__((address_space(3)))

typedef int v4i __attribute__((ext_vector_type(4)));

#if defined(__HIP_DEVICE_COMPILE__) && __has_builtin(__builtin_amdgcn_global_load_async_to_lds_b128)
#define HAVE_ASYNC_LDS 1
#else
#define HAVE_ASYNC_LDS 0
#endif

__device__ __forceinline__ void wait_async0() {
#if defined(__HIP_DEVICE_COMPILE__)
#if __has_builtin(__builtin_amdgcn_s_wait_asynccnt)
  __builtin_amdgcn_s_wait_asynccnt(0);
#else
  asm volatile("s_wait_asynccnt 0" ::: "memory");
#endif
#endif
}

// ---------------- kernel 1: pack per-row state ----------------
// table[i] = { mu, sigma^2 = exp(output[i,1]), y, bitcast(task_id) }
__global__ __launch_bounds__(TPB) void prl_prep(const float* __restrict__ out2,
                                                const float* __restrict__ y,
                                                const int*   __restrict__ tid,
                                                float4*      __restrict__ tbl) {
  int i = blockIdx.x * TPB + threadIdx.x;
  if (i < N) {
    float mu = out2[2 * i];
    float s2 = expf(out2[2 * i + 1]);   // sigma^2; sqrt never materialized
    tbl[i] = make_float4(mu, s2, y[i], __int_as_float(tid[i]));
  }
}

// ---------------- kernel 2: all-pairs loss, one wave per row ----------------
__global__ __launch_bounds__(TPB) void prl_pair(const float4* __restrict__ tbl,
                                                float*        __restrict__ per_i) {
  __shared__ float4 tile[N];   // 96 KB of the 320 KB/WGP LDS

  // Async-stage the whole table into LDS (24 x b128 per thread).
  for (int k = threadIdx.x; k < N; k += TPB) {
#if HAVE_ASYNC_LDS
    __builtin_amdgcn_global_load_async_to_lds_b128(
        (GAS v4i*)(tbl + k), (LAS v4i*)(tile + k), 0, 0);
#else
    tile[k] = tbl[k];
#endif
  }
  wait_async0();        // ASYNCcnt is per-wave; barrier publishes LDS to the WG
  __syncthreads();

  const int lane = threadIdx.x & 31;
  const int wid  = threadIdx.x >> 5;               // 8 waves per block
  const int i    = blockIdx.x * (TPB / 32) + wid;  // row owned by this wave

  const float4 me    = tile[i];
  const float  mu_i  = me.x, s2_i = me.y, y_i = me.z;
  const int    tid_i = __float_as_int(me.w);
  const float  inv_sqrt2 = 0.70710678118654752440f;

  float sum = 0.0f, cnt = 0.0f;
  const int jn = N / SPLIT;                        // 192 iterations per lane
#pragma unroll 8
  for (int jj = 0; jj < jn; ++jj) {
    // Interleaved: lanes read 32 consecutive float4 -> conflict-free DS b128.
    const float4 o = tile[(jj << 5) | lane];
    const float ydiff = o.z - y_i;
    const bool  valid = (__float_as_int(o.w) == tid_i) && (ydiff != 0.0f);
    float z = (o.x - mu_i) * rsqrtf(s2_i + o.y) * inv_sqrt2;
    z = (ydiff > 0.0f) ? z : -z;                   // erf is odd: folds pos/neg branch
    const float q    = fmaf(0.5f, erff(z), 0.5f + 1e-8f);
    const float term = -__logf(q);                 // native v_log_f32 path
    if (valid) { sum += term; cnt += 1.0f; }
  }

  // Full-wave reduction of (sum, cnt) — wave32 shuffles.
  sum += __shfl_xor(sum,  1, 32);  cnt += __shfl_xor(cnt,  1, 32);
  sum += __shfl_xor(sum,  2, 32);  cnt += __shfl_xor(cnt,  2, 32);
  sum += __shfl_xor(sum,  4, 32);  cnt += __shfl_xor(cnt,  4, 32);
  sum += __shfl_xor(sum,  8, 32);  cnt += __shfl_xor(cnt,  8, 32);
  sum += __shfl_xor(sum, 16, 32);  cnt += __shfl_xor(cnt, 16, 32);
  if (lane == 0) per_i[i] = sum / cnt;  // per-row mean (matches reference, incl. 0-count NaN)
}

// ---------------- kernel 3: deterministic mean over rows ----------------
__global__ __launch_bounds__(TPB) void prl_reduce(const float* __restrict__ per_i,
                                                  float*       __restrict__ out) {
  __shared__ float s[TPB];
  float a = 0.0f;
  for (int k = threadIdx.x; k < N; k += TPB) a += per_i[k];
  s[threadIdx.x] = a;
  __syncthreads();
  for (int w = TPB / 2; w > 0; w >>= 1) {
    if (threadIdx.x < w) s[threadIdx.x] += s[threadIdx.x + w];
    __syncthreads();
  }
  if (threadIdx.x == 0) out[0] = s[0] * (1.0f / (float)N);
}

extern "C" void kernel_launch(void* const* d_in, const int* in_sizes, int n_in,
                              void* d_out, int out_size, void* d_ws, size_t ws_size,
                              hipStream_t stream) {
  (void)in_sizes; (void)n_in; (void)out_size; (void)ws_size;
  const float* out2 = (const float*)d_in[0];   // (6144, 2) f32
  const float* y    = (const float*)d_in[1];   // (6144,)  f32
  const int*   tid  = (const int*)d_in[2];     // (6144,1) i32

  float4* tbl   = (float4*)d_ws;                                      // 96 KB
  float*  per_i = (float*)((char*)d_ws + (size_t)N * sizeof(float4)); // 24 KB

  prl_prep  <<<(N + TPB - 1) / TPB, TPB, 0, stream>>>(out2, y, tid, tbl);
  prl_pair  <<<(N * SPLIT) / TPB,   TPB, 0, stream>>>(tbl, per_i);    // 768 blocks
  prl_reduce<<<1,                   TPB, 0, stream>>>(per_i, (float*)d_out);
}